// Cost_Volume_87634512708293
// MI455X (gfx1250) — compile-verified
//
#include <hip/hip_runtime.h>

typedef float v4f __attribute__((ext_vector_type(4)));
typedef int   v4i __attribute__((ext_vector_type(4)));

// Problem geometry (fixed by the reference).
constexpr int B = 4, C = 32, H = 64, W = 128, D = 48;
constexpr int ROWS = B * C * H;          // 8192 (b,c,h) rows
constexpr int ROWS_PER_BLOCK = 8;        // 8 wave32 per block
constexpr int PAD = 48;                  // zero pad: shifted reads fall into zeros
constexpr int SROW = PAD + W;            // 176 floats per LDS row

__global__ __launch_bounds__(32 * ROWS_PER_BLOCK)
void cost_volume_kernel(const float* __restrict__ ref,
                        const float* __restrict__ tgt,
                        float* __restrict__ out) {
    __shared__ float sm[ROWS_PER_BLOCK][SROW];   // 5632 bytes

    const int lane = threadIdx.x;                 // 0..31 (wave32)
    const int ry   = threadIdx.y;                 // 0..7
    const int row  = blockIdx.x * ROWS_PER_BLOCK + ry;  // (b*C + c)*H + h

    const int h = row & (H - 1);
    const int c = (row >> 6) & (C - 1);
    const int b = row >> 11;

    const float* tgtRow = tgt + row * W;
    float* smRow = sm[ry];

    // Zero the pad region [0, PAD): lanes 0..11 write one float4 each.
    if (lane < PAD / 4) {
        v4f z = {0.f, 0.f, 0.f, 0.f};
        *(v4f*)(smRow + lane * 4) = z;
    }

    // Stage the tgt row into LDS at [PAD, PAD+W): CDNA5 async global->LDS
    // path (GLOBAL_LOAD_ASYNC_TO_LDS_B128, tracked by ASYNCcnt).
#if __has_builtin(__builtin_amdgcn_global_load_async_to_lds_b128)
    __builtin_amdgcn_global_load_async_to_lds_b128(
        (v4i __attribute__((address_space(1)))*)tgtRow + lane,
        (v4i __attribute__((address_space(3)))*)(smRow + PAD) + lane,
        0, 0);
#if __has_builtin(__builtin_amdgcn_s_wait_asynccnt)
    __builtin_amdgcn_s_wait_asynccnt(0);
#else
    asm volatile("s_wait_asynccnt 0" ::: "memory");
#endif
#else
    *(v4f*)(smRow + PAD + lane * 4) = *(const v4f*)(tgtRow + lane * 4);
#endif

    // ref row: each element read exactly once, kept in VGPRs for all 48 d.
    const v4f rv = *(const v4f*)(ref + row * W + lane * 4);

    __syncthreads();   // cross-lane visibility of pad + staged row

    const int w0       = lane * 4;
    const int hw       = H * W;          // 8192
    const int chStride = D * hw;         // 393216
    // out[b, ch, d, h, w] = (((b*2C + ch)*D + d)*H + h)*W + w
    float* outRef = out + (b * 2 * C + c) * chStride + h * W + w0;
    float* outTgt = outRef + C * chStride;           // channel c + C
    const float* smr = smRow + PAD + w0;

#pragma unroll 4
    for (int d = 0; d < D; ++d) {
        // tgt half: shifted read; mask is implicit via the zero pad.
        v4f t;
        t.x = smr[0 - d];
        t.y = smr[1 - d];
        t.z = smr[2 - d];
        t.w = smr[3 - d];
        // ref half: per-element mask (w >= d).
        v4f r;
        r.x = (w0 + 0 >= d) ? rv.x : 0.f;
        r.y = (w0 + 1 >= d) ? rv.y : 0.f;
        r.z = (w0 + 2 >= d) ? rv.z : 0.f;
        r.w = (w0 + 3 >= d) ? rv.w : 0.f;
        // Streaming write-once output: non-temporal b128 stores.
        __builtin_nontemporal_store(r, (v4f*)outRef);
        __builtin_nontemporal_store(t, (v4f*)outTgt);
        outRef += hw;
        outTgt += hw;
    }
}

extern "C" void kernel_launch(void* const* d_in, const int* in_sizes, int n_in,
                              void* d_out, int out_size, void* d_ws, size_t ws_size,
                              hipStream_t stream) {
    (void)in_sizes; (void)n_in; (void)out_size; (void)d_ws; (void)ws_size;
    const float* ref = (const float*)d_in[0];
    const float* tgt = (const float*)d_in[1];
    float* out = (float*)d_out;

    dim3 block(32, ROWS_PER_BLOCK);          // 8 wave32 per WGP-friendly block
    dim3 grid(ROWS / ROWS_PER_BLOCK);        // 1024 blocks
    cost_volume_kernel<<<grid, block, 0, stream>>>(ref, tgt, out);
}